// SeparationEmbedding_42554535969388
// MI455X (gfx1250) — compile-verified
//
#include <hip/hip_runtime.h>
#include <hip/hip_bf16.h>

typedef __attribute__((ext_vector_type(4))) unsigned int u32x4;
typedef __attribute__((ext_vector_type(4))) int          i32x4;
typedef __attribute__((ext_vector_type(8))) int          i32x8;
typedef __attribute__((ext_vector_type(4))) float        f32x4;

#define ROWS 11
#define COLS 64
#define TABN (ROWS * COLS)   // 704 floats = 2816 bytes

__device__ __forceinline__ int bin_code(int x) {
    // searchsorted(BINS, x, side='left') == count of bins strictly < x
    return (x > 1) + (x > 2) + (x > 3) + (x > 4) + (x > 5)
         + (x > 10) + (x > 15) + (x > 20) + (x > 30) + (x > 50);
}

__global__ __launch_bounds__(256) void SeparationEmbedding_kernel(
    const long long* __restrict__ edge,   // [2*E] int64: row0 = i, row1 = j
    const float*     __restrict__ emb,    // [11*64] f32
    float*           __restrict__ out,    // [E*64] f32
    long long E)
{
    // Embedding table staged in LDS; single static LDS object -> offset 0.
    __shared__ __align__(16) float ltab[TABN];

    // ---- Tensor Data Mover: DMA the 11x64 f32 table (2816 B) global -> LDS ----
    if (threadIdx.x < 32) {   // wave 0 of the workgroup issues the TDM op
        unsigned long long ga = (unsigned long long)emb;
        u32x4 g0;
        g0[0] = 1u;                                        // count=1 (valid user D#)
        g0[1] = 0u;                                        // lds_addr = 0
        g0[2] = (unsigned)(ga & 0xffffffffull);            // global_addr[31:0]
        g0[3] = (unsigned)((ga >> 32) & 0x01ffffffull)     // global_addr[56:32]
              | (2u << 30);                                // type = 2 ("image")
        i32x8 g1;
        g1[0] = (int)(2u << 16);                           // data_size = 2 -> 4 bytes
        g1[1] = (int)((unsigned)COLS << 16);               // tensor_dim0 = 64 (bits 79:48)
        g1[2] = (int)((unsigned)ROWS << 16);               // tensor_dim1 = 11 (bits 111:80)
        g1[3] = (int)((unsigned)COLS << 16);               // tile_dim0   = 64 (bits 127:112)
        g1[4] = ROWS;                                      // tile_dim1   = 11
        g1[5] = COLS;                                      // tensor_dim0_stride = 64
        g1[6] = 0;
        g1[7] = 0;
        i32x4 z4 = {0, 0, 0, 0};                           // groups 2/3 unused (2D tile)
        i32x8 z8 = {0, 0, 0, 0, 0, 0, 0, 0};
        __builtin_amdgcn_tensor_load_to_lds(g0, g1, z4, z4, z8, 0);
        __builtin_amdgcn_s_wait_tensorcnt(0);
    }
    __syncthreads();

    // Authoritative fill (last writer): guarantees table contents and makes the
    // LDS object visibly initialized to the compiler so the gather loop is not
    // folded away (TDM's LDS write is opaque to LLVM alias analysis).
    for (int t = threadIdx.x; t < TABN; t += blockDim.x)
        ltab[t] = emb[t];
    __syncthreads();

    const int lane = threadIdx.x & 31;
    const int half = lane >> 4;        // 0: edge 2k, 1: edge 2k+1
    const int col  = lane & 15;        // f32x4 column within a 64-float row

    const long long wid    = ((long long)blockIdx.x * blockDim.x + threadIdx.x) >> 5;
    const long long nwaves = ((long long)gridDim.x * blockDim.x) >> 5;
    const f32x4* lt  = (const f32x4*)ltab;   // 11 rows x 16 f32x4
    const f32x4* ltp = lt + col;             // this lane's column slice

    for (long long base = wid * 32; base < E; base += nwaves * 32) {
        const bool full = (base + 32 <= E);

        // Phase 1: each lane computes the bin code for one edge (coalesced b64 loads).
        long long e = base + lane;
        int code = 0;
        if (full || e < E) {
            long long i = __builtin_nontemporal_load(&edge[e]);
            long long j = __builtin_nontemporal_load(&edge[E + e]);
            long long s = i - j;
            code = bin_code((int)(s < 0 ? -s : s));   // |sep| < 100000, fits i32
        }

        // Phase 2: 16 lanes per edge, each writes one f32x4 -> 256B contiguous
        // per edge, b128 non-temporal stores (output is 1 GB streaming).
        // f32x4-unit address: (base + 2k + half)*16 + col == op + k*32
        f32x4* op = (f32x4*)(out + base * 64) + (half * 16 + col);
        if (full) {
            #pragma unroll
            for (int k = 0; k < 16; ++k) {
                int c = __shfl(code, 2 * k + half, 32);
                __builtin_nontemporal_store(ltp[c * 16], op + k * 32);
            }
        } else {
            #pragma unroll
            for (int k = 0; k < 16; ++k) {
                int src = 2 * k + half;
                int c = __shfl(code, src, 32);
                if (base + src < E)
                    __builtin_nontemporal_store(ltp[c * 16], op + k * 32);
            }
        }
    }
}

extern "C" void kernel_launch(void* const* d_in, const int* in_sizes, int n_in,
                              void* d_out, int out_size, void* d_ws, size_t ws_size,
                              hipStream_t stream) {
    const long long* edge = (const long long*)d_in[0];  // int64 [2, E]
    const float*     emb  = (const float*)d_in[1];      // f32 [11, 64]
    float*           out  = (float*)d_out;              // f32 [E, 64]

    long long E = (long long)in_sizes[0] / 2;
    long long waves  = (E + 31) / 32;          // 32 edges per wave-iteration
    long long blocks = (waves + 7) / 8;        // 8 wave32 per 256-thread block
    if (blocks < 1) blocks = 1;

    SeparationEmbedding_kernel<<<(int)blocks, 256, 0, stream>>>(edge, emb, out, E);
}